// WavAnalysis_62672162783937
// MI455X (gfx1250) — compile-verified
//
#include <hip/hip_runtime.h>

// Fused 4-scale undecimated starlet transform for MI455X (gfx1250).
//
// Per workgroup: stage a 124x124 (64x64 tile + 30px symmetric halo) region
// into LDS via GLOBAL_LOAD_ASYNC_TO_LDS_B32 (ASYNCcnt path), then 4 scales
// of separable dilated 5-tap convolution entirely in LDS; the 4 normalized
// detail bands stream to HBM with non-temporal b64 stores.
//
// - Stride-d output pairing: the 5-tap windows of outputs (p, p+d) share
//   4 taps -> 6 b64 LDS loads produce 4 outputs (2x fewer LDS dwords than
//   naive; LDS banks are the binding resource at 64 dwords/clk/WGP).
// - Persistent over 4 batches per WG with a dedicated stage buffer: the
//   next batch's async load is issued right after scale 0 releases sS and
//   drains behind scales 1-3 (fixed ping-pong S->A->B->A->B).
// - LDS: 4 x 124x124 f32 = 246 KB (ISA allows 320 KB/WG; >160 KB verified
//   to compile in round 3). One WG/WGP -> 512 threads (16 wave32).

#define TW 64
#define TH 64
#define HALO 30
#define RW (TW + 2 * HALO) /* 124 */
#define RH (TH + 2 * HALO) /* 124 */
#define NTHREADS 512
#define IMG_H 1024
#define IMG_W 1024
#define NBATCH 16
#define BPW 4 /* batches per workgroup */
#define NSCALES 4

typedef float v2f __attribute__((ext_vector_type(2)));

#if __has_builtin(__builtin_amdgcn_global_load_async_to_lds_b32) && \
    __has_builtin(__builtin_amdgcn_s_wait_asynccnt)
#define USE_ASYNC_LDS 1
#define ASYNC_WAIT() __builtin_amdgcn_s_wait_asynccnt(0)
typedef __attribute__((address_space(1))) int gint_t;
typedef __attribute__((address_space(3))) int lint_t;
#else
#define USE_ASYNC_LDS 0
#define ASYNC_WAIT()
#endif

// radius of valid region of low_s around the tile: r_s = 2*sum_{j>=s} 2^j
__host__ __device__ constexpr int RADC(int s) {
    return 2 * ((1 << NSCALES) - (1 << s));
}

__device__ __forceinline__ int reflect_idx(int i, int n) {
    // jnp.pad mode='symmetric': -1 -> 0, -2 -> 1, n -> n-1, n+1 -> n-2
    i = (i < 0) ? (-i - 1) : i;
    i = (i >= n) ? (2 * n - 1 - i) : i;
    return i;
}

__device__ __forceinline__ void stage_tile(const float* __restrict__ img, int b,
                                           int tx0, int ty0,
                                           float* __restrict__ dstLds, int tid) {
    const float* imgb = img + (size_t)b * IMG_H * IMG_W;
    for (int idx = tid; idx < RH * RW; idx += NTHREADS) {
        const int ry = idx / RW;
        const int rx = idx - ry * RW;
        const int gy = reflect_idx(ty0 + ry - HALO, IMG_H);
        const int gx = reflect_idx(tx0 + rx - HALO, IMG_W);
        const float* src = imgb + (size_t)gy * IMG_W + gx;
#if USE_ASYNC_LDS
        __builtin_amdgcn_global_load_async_to_lds_b32(
            (gint_t*)(void*)src, (lint_t*)(dstLds + idx), 0, 0);
#else
        dstLds[idx] = *src;
#endif
    }
}

// Horizontal 5-tap dilated pass: cur -> tmp over rows of R_s, cols of R_{s+1}.
// Each item produces 4 outputs: float2 pair at xb and float2 pair at xb+d.
template <int SC>
__device__ __forceinline__ void hpass(const float* __restrict__ cur,
                                      float* __restrict__ tmp, int tid) {
    constexpr int d    = 1 << SC;
    constexpr int ri   = RADC(SC);
    constexpr int rn   = RADC(SC + 1);
    constexpr int rows = TH + 2 * ri;
    constexpr int cols = TW + 2 * rn;
    constexpr int y0 = HALO - ri, x0 = HALO - rn;
    constexpr int colI = cols / 4;
    for (int idx = tid; idx < rows * colI; idx += NTHREADS) {
        const int yy = idx / colI;
        const int i  = idx - yy * colI;
        const int rowbase = (y0 + yy) * RW + x0;
        if constexpr (d == 1) {
            // outputs xb..xb+3; taps span xb-2..xb+5 -> 4 aligned b64 loads
            const int xb = i * 4;
            const float* p = cur + rowbase + xb;
            v2f a  = *(const v2f*)(p - 2);
            v2f bb = *(const v2f*)(p);
            v2f c  = *(const v2f*)(p + 2);
            v2f e  = *(const v2f*)(p + 4);
            v2f r0, r1;
            r0.x = 0.0625f * (a.x + c.x)  + 0.25f * (a.y + bb.y) + 0.375f * bb.x;
            r0.y = 0.0625f * (a.y + c.y)  + 0.25f * (bb.x + c.x) + 0.375f * bb.y;
            r1.x = 0.0625f * (bb.x + e.x) + 0.25f * (bb.y + c.y) + 0.375f * c.x;
            r1.y = 0.0625f * (bb.y + e.y) + 0.25f * (c.x + e.x)  + 0.375f * c.y;
            *(v2f*)(tmp + rowbase + xb)     = r0;
            *(v2f*)(tmp + rowbase + xb + 2) = r1;
        } else {
            // stride-d pairing: outputs {xb,xb+1} and {xb+d,xb+d+1}
            // share 4 of 5 taps -> 6 b64 loads / 4 outputs
            constexpr int hd = d / 2;
            const int blk = i / hd;
            const int ln  = i - blk * hd;
            const int xb  = blk * 2 * d + ln * 2;
            const float* p = cur + rowbase + xb;
            v2f m2 = *(const v2f*)(p - 2 * d);
            v2f m1 = *(const v2f*)(p - d);
            v2f c0 = *(const v2f*)(p);
            v2f p1 = *(const v2f*)(p + d);
            v2f p2 = *(const v2f*)(p + 2 * d);
            v2f p3 = *(const v2f*)(p + 3 * d);
            v2f r0, r1;
            r0.x = 0.0625f * (m2.x + p2.x) + 0.25f * (m1.x + p1.x) + 0.375f * c0.x;
            r0.y = 0.0625f * (m2.y + p2.y) + 0.25f * (m1.y + p1.y) + 0.375f * c0.y;
            r1.x = 0.0625f * (m1.x + p3.x) + 0.25f * (c0.x + p2.x) + 0.375f * p1.x;
            r1.y = 0.0625f * (m1.y + p3.y) + 0.25f * (c0.y + p2.y) + 0.375f * p1.y;
            *(v2f*)(tmp + rowbase + xb)     = r0;
            *(v2f*)(tmp + rowbase + xb + d) = r1;
        }
    }
}

// Vertical 5-tap dilated pass: tmp -> nxt over R_{s+1}; fused detail output
// w = (cur - nxt) * invn. Each item: rows (yb, yb+d) x float2 cols.
template <int SC>
__device__ __forceinline__ void vpass(const float* __restrict__ tmp,
                                      const float* __restrict__ cur,
                                      float* __restrict__ nxt,
                                      float* __restrict__ out, float invn,
                                      int b, int tx0, int ty0, int tid) {
    constexpr int d    = 1 << SC;
    constexpr int rn   = RADC(SC + 1);
    constexpr int rows = TH + 2 * rn;
    constexpr int cols = TW + 2 * rn;
    constexpr int y0 = HALO - rn, x0 = HALO - rn;
    constexpr int rowI = rows / 2;
    constexpr int colI = cols / 2;
    for (int idx = tid; idx < rowI * colI; idx += NTHREADS) {
        const int j = idx / colI;
        const int i = idx - j * colI;
        const int blk = j / d;
        const int r   = j - blk * d;
        const int yy  = blk * 2 * d + r;
        const int ry  = y0 + yy;
        const int rx  = x0 + i * 2;
        const int base = ry * RW + rx;
        v2f m2 = *(const v2f*)(tmp + base - 2 * d * RW);
        v2f m1 = *(const v2f*)(tmp + base - d * RW);
        v2f c0 = *(const v2f*)(tmp + base);
        v2f p1 = *(const v2f*)(tmp + base + d * RW);
        v2f p2 = *(const v2f*)(tmp + base + 2 * d * RW);
        v2f p3 = *(const v2f*)(tmp + base + 3 * d * RW);
        v2f nl0, nl1;
        nl0.x = 0.0625f * (m2.x + p2.x) + 0.25f * (m1.x + p1.x) + 0.375f * c0.x;
        nl0.y = 0.0625f * (m2.y + p2.y) + 0.25f * (m1.y + p1.y) + 0.375f * c0.y;
        nl1.x = 0.0625f * (m1.x + p3.x) + 0.25f * (c0.x + p2.x) + 0.375f * p1.x;
        nl1.y = 0.0625f * (m1.y + p3.y) + 0.25f * (c0.y + p2.y) + 0.375f * p1.y;
        *(v2f*)(nxt + base)            = nl0;
        *(v2f*)(nxt + base + d * RW)   = nl1;

        const int ox = rx - HALO;
        if ((unsigned)ox < (unsigned)TW) {
            const int oy0 = ry - HALO;
            if ((unsigned)oy0 < (unsigned)TH) {
                v2f c = *(const v2f*)(cur + base);
                v2f w;
                w.x = (c.x - nl0.x) * invn;
                w.y = (c.y - nl0.y) * invn;
                float* dst = out + ((((size_t)SC * NBATCH + b) * IMG_H +
                                     (size_t)(ty0 + oy0)) * IMG_W + (tx0 + ox));
                __builtin_nontemporal_store(w, (v2f*)dst);
            }
            const int oy1 = ry + d - HALO;
            if ((unsigned)oy1 < (unsigned)TH) {
                v2f c = *(const v2f*)(cur + base + d * RW);
                v2f w;
                w.x = (c.x - nl1.x) * invn;
                w.y = (c.y - nl1.y) * invn;
                float* dst = out + ((((size_t)SC * NBATCH + b) * IMG_H +
                                     (size_t)(ty0 + oy1)) * IMG_W + (tx0 + ox));
                __builtin_nontemporal_store(w, (v2f*)dst);
            }
        }
    }
}

__global__ __launch_bounds__(NTHREADS) void starlet_fused_kernel(
    const float* __restrict__ img, const float* __restrict__ wnorm,
    float* __restrict__ out) {
    __shared__ float sS[RH * RW];  // stage-in buffer (scale 0 input)
    __shared__ float sA[RH * RW];
    __shared__ float sB[RH * RW];
    __shared__ float sT[RH * RW];  // horizontal-pass scratch

    const int tid = threadIdx.x;
    const int tx0 = blockIdx.x * TW;
    const int ty0 = blockIdx.y * TH;
    const int b0  = blockIdx.z * BPW;

    float invn[NSCALES];
#pragma unroll
    for (int s = 0; s < NSCALES; ++s) invn[s] = 1.0f / wnorm[s];

    stage_tile(img, b0, tx0, ty0, sS, tid);
    ASYNC_WAIT();
    __syncthreads();

#pragma unroll 1
    for (int bi = 0; bi < BPW; ++bi) {
        const int b = b0 + bi;

        // scale 0: S -> A (S is free afterwards -> prefetch next batch)
        hpass<0>(sS, sT, tid);
        __syncthreads();
        vpass<0>(sT, sS, sA, out, invn[0], b, tx0, ty0, tid);
        __syncthreads();
        if (bi + 1 < BPW) stage_tile(img, b + 1, tx0, ty0, sS, tid);

        // scale 1: A -> B
        hpass<1>(sA, sT, tid);
        __syncthreads();
        vpass<1>(sT, sA, sB, out, invn[1], b, tx0, ty0, tid);
        __syncthreads();

        // scale 2: B -> A
        hpass<2>(sB, sT, tid);
        __syncthreads();
        vpass<2>(sT, sB, sA, out, invn[2], b, tx0, ty0, tid);
        __syncthreads();

        // scale 3: A -> B
        hpass<3>(sA, sT, tid);
        __syncthreads();
        vpass<3>(sT, sA, sB, out, invn[3], b, tx0, ty0, tid);
        __syncthreads();

        // next batch's stage must have landed before scale 0 reads sS
        ASYNC_WAIT();
        __syncthreads();
    }
}

extern "C" void kernel_launch(void* const* d_in, const int* in_sizes, int n_in,
                              void* d_out, int out_size, void* d_ws, size_t ws_size,
                              hipStream_t stream) {
    (void)in_sizes; (void)n_in; (void)out_size; (void)d_ws; (void)ws_size;
    const float* img = (const float*)d_in[0];  // [16,1024,1024,1] f32
    const float* wn  = (const float*)d_in[1];  // [4] f32
    float* out = (float*)d_out;                // [4,16,1024,1024,1] f32

    dim3 grid(IMG_W / TW, IMG_H / TH, NBATCH / BPW);  // (16, 16, 4)
    starlet_fused_kernel<<<grid, NTHREADS, 0, stream>>>(img, wn, out);
}